// _Node_89292370084147
// MI455X (gfx1250) — compile-verified
//
#include <hip/hip_runtime.h>

// out = sin(s)*cos(x) + exp(-y^2)*tanh(s) + log(|s|+1)/pi,  s = a*x + b*y
// Pure streaming elementwise op: HBM-bound (12 B/elem, 402 MB total).
// Strategy: float4 (B128) non-temporal loads/stores, native gfx1250
// transcendentals incl. v_tanh_f32 via inline asm.

typedef float f32x4 __attribute__((ext_vector_type(4)));

#define INV_PI 0.3183098861837907f

__device__ __forceinline__ float fused_point(float xx, float yy, float a, float b) {
    float s = __builtin_fmaf(a, xx, b * yy);
    // native CDNA5 tanh (TRANS32 unit) — avoids ocml polynomial/div expansion
    float th;
    asm("v_tanh_f32 %0, %1" : "=v"(th) : "v"(s));
    float r = __sinf(s) * __cosf(xx);
    r = __builtin_fmaf(__expf(-yy * yy), th, r);
    r = __builtin_fmaf(__logf(__builtin_fabsf(s) + 1.0f), INV_PI, r);
    return r;
}

__global__ void fused_expr_v4(const float* __restrict__ x,
                              const float* __restrict__ y,
                              const float* __restrict__ pa,
                              const float* __restrict__ pb,
                              float* __restrict__ out,
                              long long n4) {
    long long i = (long long)blockIdx.x * blockDim.x + threadIdx.x;
    if (i >= n4) return;

    const float a = pa[0];   // uniform -> scalar load, L2 resident
    const float b = pb[0];

    // 128-bit streaming loads, non-temporal (dataset > L2, zero reuse)
    f32x4 xv = __builtin_nontemporal_load((const f32x4*)x + i);
    f32x4 yv = __builtin_nontemporal_load((const f32x4*)y + i);

    f32x4 ov;
#pragma unroll
    for (int k = 0; k < 4; ++k) {
        ov[k] = fused_point(xv[k], yv[k], a, b);
    }

    __builtin_nontemporal_store(ov, (f32x4*)out + i);
}

// scalar tail for n % 4 != 0 (not hit for 32*1024*1024, kept for generality)
__global__ void fused_expr_tail(const float* __restrict__ x,
                                const float* __restrict__ y,
                                const float* __restrict__ pa,
                                const float* __restrict__ pb,
                                float* __restrict__ out,
                                long long start, long long n) {
    long long i = start + threadIdx.x;
    if (i >= n) return;
    out[i] = fused_point(x[i], y[i], pa[0], pb[0]);
}

extern "C" void kernel_launch(void* const* d_in, const int* in_sizes, int n_in,
                              void* d_out, int out_size, void* d_ws, size_t ws_size,
                              hipStream_t stream) {
    (void)in_sizes; (void)n_in; (void)d_ws; (void)ws_size;

    const float* x  = (const float*)d_in[0];
    const float* y  = (const float*)d_in[1];
    const float* pa = (const float*)d_in[2];
    const float* pb = (const float*)d_in[3];
    float* out      = (float*)d_out;

    const long long n  = (long long)out_size;
    const long long n4 = n >> 2;                 // float4 chunks
    const long long rem_start = n4 << 2;

    if (n4 > 0) {
        const int block = 256;                   // 8 wave32 per block
        const long long grid = (n4 + block - 1) / block;
        fused_expr_v4<<<(unsigned int)grid, block, 0, stream>>>(x, y, pa, pb, out, n4);
    }
    if (rem_start < n) {
        fused_expr_tail<<<1, 64, 0, stream>>>(x, y, pa, pb, out, rem_start, n);
    }
}